// RMM_71124658422342
// MI455X (gfx1250) — compile-verified
//
#include <hip/hip_runtime.h>

typedef float v2f __attribute__((ext_vector_type(2)));
typedef float v8f __attribute__((ext_vector_type(8)));

#define N_C 1024          // centroids
#define M_F 64            // features
#define ROWS_PER_WG 32
#define WAVES 8
#define A_STRIDE 132      // 128 + pad -> conflict-free ds_load_b64 fragment reads

__global__ __launch_bounds__(256)
void gmm_trunc_softmax_kernel(const float* __restrict__ samples,
                              const float* __restrict__ mask,
                              const float* __restrict__ cen,
                              const float* __restrict__ sd,
                              const float* __restrict__ lmarg,
                              float* __restrict__ out_pdf,
                              float* __restrict__ out_lpdf)
{
  __shared__ float A_lds[ROWS_PER_WG][A_STRIDE]; // [r][k]: k<64 -> q, k>=64 -> q*a
  __shared__ float Ad_lds[ROWS_PER_WG];          // per-row sum q*a*a
  __shared__ float lp_lds[N_C];                  // log_softmax(log_marginal)
  __shared__ float red[256];
  __shared__ float part[ROWS_PER_WG][WAVES];
  __shared__ float rowred[ROWS_PER_WG];

  const int tid   = threadIdx.x;
  const int lane  = tid & 31;
  const int wave  = tid >> 5;
  const int hl    = lane >> 4;       // half of wave (0: lanes 0-15, 1: 16-31)
  const int ml    = lane & 15;
  const int wgRow = blockIdx.x * ROWS_PER_WG;
  const int wvCol = wave * 128;

  if (tid < ROWS_PER_WG) Ad_lds[tid] = 0.f;
  __syncthreads();

  // ---------- Phase 0a: fused A operand  [ sq_mask | sq_mask*A ]  + A_d ----------
  for (int i = tid; i < ROWS_PER_WG * M_F; i += 256) {
    int r = i >> 6, k = i & 63;
    float a  = samples[(size_t)(wgRow + r) * M_F + k];
    float mk = mask   [(size_t)(wgRow + r) * M_F + k];
    float m  = mk / sd[k];
    float q  = m * m;
    A_lds[r][k]      = q;
    A_lds[r][64 + k] = q * a;
    atomicAdd(&Ad_lds[r], q * a * a);    // ds_add_f32
  }

  // ---------- Phase 0b: log_softmax(log_marginal) ----------
  float pm = -3.402823466e38f;
  for (int j = tid; j < N_C; j += 256) { float v = lmarg[j]; lp_lds[j] = v; pm = fmaxf(pm, v); }
  red[tid] = pm; __syncthreads();
  for (int s = 128; s > 0; s >>= 1) { if (tid < s) red[tid] = fmaxf(red[tid], red[tid + s]); __syncthreads(); }
  float lmax = red[0];
  __syncthreads();
  float ps = 0.f;
  for (int j = tid; j < N_C; j += 256) ps += __expf(lp_lds[j] - lmax);
  red[tid] = ps; __syncthreads();
  for (int s = 128; s > 0; s >>= 1) { if (tid < s) red[tid] += red[tid + s]; __syncthreads(); }
  float lz = lmax + __logf(red[0]);
  __syncthreads();
  for (int j = tid; j < N_C; j += 256) lp_lds[j] -= lz;
  __syncthreads();

  // ---------- Phase 1: K=128 GEMM via V_WMMA_F32_16X16X4_F32 ----------
  v8f acc[2][8];
  #pragma unroll
  for (int rt = 0; rt < 2; ++rt)
    #pragma unroll
    for (int t = 0; t < 8; ++t) { v8f z = {0.f,0.f,0.f,0.f,0.f,0.f,0.f,0.f}; acc[rt][t] = z; }

  int jbase[8];
  #pragma unroll
  for (int t = 0; t < 8; ++t) jbase[t] = (wvCol + t * 16 + ml) * M_F;

  const int klane = 2 * hl;   // ISA A/B frag layout: upper half-lanes carry K+2,K+3

  for (int k0 = 0; k0 < 64; k0 += 4) {
    v2f aq0 = *(const v2f*)&A_lds[ml     ][     k0 + klane];  // rows 0-15, q part
    v2f aq1 = *(const v2f*)&A_lds[16 + ml][     k0 + klane];  // rows 16-31, q part
    v2f al0 = *(const v2f*)&A_lds[ml     ][64 + k0 + klane];  // rows 0-15, q*a part
    v2f al1 = *(const v2f*)&A_lds[16 + ml][64 + k0 + klane];  // rows 16-31, q*a part
    #pragma unroll
    for (int t = 0; t < 8; ++t) {
      v2f c2 = *(const v2f*)(cen + jbase[t] + k0 + klane);    // one load feeds both halves
      v2f bq; bq.x = c2.x * c2.x;  bq.y = c2.y * c2.y;        // B[k<64]  =  C^2
      v2f bl; bl.x = -2.f * c2.x;  bl.y = -2.f * c2.y;        // B[k>=64] = -2C
      acc[0][t] = __builtin_amdgcn_wmma_f32_16x16x4_f32(false, aq0, false, bq, (short)0, acc[0][t], false, false);
      acc[1][t] = __builtin_amdgcn_wmma_f32_16x16x4_f32(false, aq1, false, bq, (short)0, acc[1][t], false, false);
      acc[0][t] = __builtin_amdgcn_wmma_f32_16x16x4_f32(false, al0, false, bl, (short)0, acc[0][t], false, false);
      acc[1][t] = __builtin_amdgcn_wmma_f32_16x16x4_f32(false, al1, false, bl, (short)0, acc[1][t], false, false);
    }
  }

  // ---------- Phase 2: logits = log_prior - 0.5*relu(A_d + acc) ----------
  float lpv[8];
  #pragma unroll
  for (int t = 0; t < 8; ++t) lpv[t] = lp_lds[wvCol + t * 16 + ml];

  #pragma unroll
  for (int rt = 0; rt < 2; ++rt)
    #pragma unroll
    for (int v = 0; v < 8; ++v) {
      int r = rt * 16 + v + 8 * hl;      // C layout: reg v = rows v (low half), v+8 (high half)
      float ad = Ad_lds[r];
      #pragma unroll
      for (int t = 0; t < 8; ++t) {
        float dist = fmaxf(0.f, ad + acc[rt][t][v]);
        acc[rt][t][v] = lpv[t] - 0.5f * dist;
      }
    }

  // ---------- Phase 3: row max (wave-local shuffle, then cross-wave via LDS) ----------
  #pragma unroll
  for (int rt = 0; rt < 2; ++rt)
    #pragma unroll
    for (int v = 0; v < 8; ++v) {
      float m = acc[rt][0][v];
      #pragma unroll
      for (int t = 1; t < 8; ++t) m = fmaxf(m, acc[rt][t][v]);
      #pragma unroll
      for (int mk = 1; mk < 16; mk <<= 1) m = fmaxf(m, __shfl_xor(m, mk, 32));
      if (ml == 0) part[rt * 16 + v + 8 * hl][wave] = m;
    }
  __syncthreads();
  if (tid < ROWS_PER_WG) {
    float m = part[tid][0];
    #pragma unroll
    for (int w = 1; w < WAVES; ++w) m = fmaxf(m, part[tid][w]);
    rowred[tid] = m;
  }
  __syncthreads();

  // ---------- Phase 4: e = exp(logit - rowmax); truncate (e > 0.05); kept-sum ----------
  #pragma unroll
  for (int rt = 0; rt < 2; ++rt)
    #pragma unroll
    for (int v = 0; v < 8; ++v) {
      int r = rt * 16 + v + 8 * hl;
      float rm = rowred[r];
      float s = 0.f;
      #pragma unroll
      for (int t = 0; t < 8; ++t) {
        float e = __expf(acc[rt][t][v] - rm);
        e = (e > 0.05f) ? e : 0.f;       // == pdf > 0.05*rowmax(pdf), s1 cancels
        acc[rt][t][v] = e;
        s += e;
      }
      #pragma unroll
      for (int mk = 1; mk < 16; mk <<= 1) s += __shfl_xor(s, mk, 32);
      if (ml == 0) part[r][wave] = s;
    }
  __syncthreads();
  if (tid < ROWS_PER_WG) {
    float s = part[tid][0];
    #pragma unroll
    for (int w = 1; w < WAVES; ++w) s += part[tid][w];
    rowred[tid] = s;
  }
  __syncthreads();

  // ---------- Phase 5: normalize + log, non-temporal streaming stores ----------
  const size_t base = (size_t)(wgRow + 8 * hl) * N_C + (wvCol + ml);
  float* p0 = out_pdf  + base;
  float* p1 = out_lpdf + base;

  #pragma unroll
  for (int rt = 0; rt < 2; ++rt)
    #pragma unroll
    for (int v = 0; v < 8; ++v) {
      int r = rt * 16 + v + 8 * hl;
      float inv = 1.0f / rowred[r];
      #pragma unroll
      for (int t = 0; t < 8; ++t) {
        float pdf = acc[rt][t][v] * inv;
        float lpd = __logf(pdf + 1e-20f);
        size_t off = (size_t)(rt * 16 + v) * N_C + (size_t)t * 16;
        __builtin_nontemporal_store(pdf, p0 + off);
        __builtin_nontemporal_store(lpd, p1 + off);
      }
    }
}

extern "C" void kernel_launch(void* const* d_in, const int* in_sizes, int n_in,
                              void* d_out, int out_size, void* d_ws, size_t ws_size,
                              hipStream_t stream) {
  const float* samples = (const float*)d_in[0];
  const float* mask    = (const float*)d_in[1];
  const float* cen     = (const float*)d_in[2];
  const float* sd      = (const float*)d_in[3];
  const float* lmarg   = (const float*)d_in[4];

  const int rowsTotal = in_sizes[0] / M_F;          // 32768
  float* out_pdf  = (float*)d_out;
  float* out_lpdf = out_pdf + (size_t)rowsTotal * N_C;

  dim3 grid(rowsTotal / ROWS_PER_WG);               // 1024 workgroups
  gmm_trunc_softmax_kernel<<<grid, 256, 0, stream>>>(
      samples, mask, cen, sd, lmarg, out_pdf, out_lpdf);
}